// YoloLoss_84215718740121
// MI455X (gfx1250) — compile-verified
//
#include <hip/hip_runtime.h>
#include <stdint.h>

// ---------------------------------------------------------------------------
// YOLO loss for MI455X (gfx1250, wave32).
// Bandwidth-bound streaming reduction; obj-mask gating skips ~2/3 of the
// preCondClasses/preBoxes traffic. CDNA5-specific paths used:
//   * global_load_async_to_lds_b32 / s_wait_asynccnt  (GT staging DMA)
//   * v_wmma_f32_16x16x4_f32 with ones-A              (deterministic reduce)
// ---------------------------------------------------------------------------

#define S_    7
#define B_    2
#define NC_   20
#define IMGF  448.0f
#define CLIPF 447.0f
#define GRIDF 64.0f
#define NGT_  20
#define ITEMS_PER_BLOCK 4
#define THREADS_ 256

typedef __attribute__((ext_vector_type(2))) float v2f;
typedef __attribute__((ext_vector_type(8))) float v8f;

__device__ __forceinline__ uint32_t lds_byte_off(const void* p) {
    // addrspace(3) pointers carry the workgroup-relative LDS byte offset in
    // the low 32 bits of the generic address (aperture lives in the high bits).
    return (uint32_t)(uintptr_t)p;
}

__device__ __forceinline__ float clip01(float v) {
    return fminf(fmaxf(v, 0.0f), CLIPF);
}

// IoU of one predicted box (normalized cx,cy,w,h) against precomputed GT
// corners, matching the reference op order exactly.
__device__ __forceinline__ float iou_box(float px, float py, float pw, float ph,
                                         float ltx, float lty,
                                         float gx1, float gy1, float gx2, float gy2,
                                         float ga) {
    float cx = px * GRIDF + ltx;
    float cy = py * GRIDF + lty;
    float hw = pw * IMGF * 0.5f;
    float hh = ph * IMGF * 0.5f;
    float x1 = clip01(cx - hw), y1 = clip01(cy - hh);
    float x2 = clip01(cx + hw), y2 = clip01(cy + hh);
    float ix = fmaxf(fminf(x2, gx2) - fmaxf(x1, gx1), 0.0f);
    float iy = fmaxf(fminf(y2, gy2) - fmaxf(y1, gy1), 0.0f);
    float inter = ix * iy;
    float a1 = (x2 - x1) * (y2 - y1);
    return inter / (a1 + ga - inter + 0.0001f);
}

__global__ __launch_bounds__(THREADS_)
void yolo_main_kernel(const float* __restrict__ conf,   // [N,7,7,2]
                      const float* __restrict__ boxes,  // [N,7,7,8]
                      const float* __restrict__ cls,    // [N,7,7,20]
                      const float* __restrict__ gt,     // [N,20,4]
                      const int*   __restrict__ lbl,    // [N,20]
                      float*       __restrict__ partial)// [nblocks,4]
{
    __shared__ float ldsGT [ITEMS_PER_BLOCK * NGT_ * 4];   // 320 dwords
    __shared__ int   ldsLB [ITEMS_PER_BLOCK * NGT_];       //  80 dwords
    __shared__ float ldsOff[ITEMS_PER_BLOCK * NGT_ * 4];
    __shared__ int   ldsFlat[ITEMS_PER_BLOCK * NGT_];
    __shared__ float4 red[THREADS_];

    const int tid = threadIdx.x;
    const int blk = blockIdx.x;

    // ---- Phase A: async DMA ground truth (+labels) into LDS --------------
    {
        const float* gGT = gt  + (size_t)blk * (ITEMS_PER_BLOCK * NGT_ * 4);
        const int*   gLB = lbl + (size_t)blk * (ITEMS_PER_BLOCK * NGT_);
        const int total = ITEMS_PER_BLOCK * NGT_ * 5;   // 400 dwords
        for (int t = tid; t < total; t += THREADS_) {
            uint64_t ga;
            uint32_t lo;
            if (t < ITEMS_PER_BLOCK * NGT_ * 4) {
                ga = (uint64_t)(uintptr_t)(gGT + t);
                lo = lds_byte_off(&ldsGT[t]);
            } else {
                int u = t - ITEMS_PER_BLOCK * NGT_ * 4;
                ga = (uint64_t)(uintptr_t)(gLB + u);
                lo = lds_byte_off(&ldsLB[u]);
            }
            asm volatile("global_load_async_to_lds_b32 %0, %1, off"
                         :: "v"(lo), "v"(ga) : "memory");
        }
        asm volatile("s_wait_asynccnt 0" ::: "memory");
        __syncthreads();
    }

    const int item   = tid >> 6;   // 0..3 : batch item within block
    const int lane64 = tid & 63;
    const int n      = blk * ITEMS_PER_BLOCK + item;

    // ---- Phase B: encode each GT box -> cell index + normalized offsets --
    if (lane64 < NGT_) {
        const int base = (item * NGT_ + lane64) * 4;
        float x1 = ldsGT[base + 0], y1 = ldsGT[base + 1];
        float x2 = ldsGT[base + 2], y2 = ldsGT[base + 3];
        float w = x2 - x1, h = y2 - y1;
        float x = x1 + w * 0.5f, y = y1 + h * 0.5f;
        int cx = (int)floorf(x / GRIDF);
        int cy = (int)floorf(y / GRIDF);
        ldsFlat[item * NGT_ + lane64] = cy * S_ + cx;
        ldsOff[base + 0] = (x - (float)cx * GRIDF) / GRIDF;
        ldsOff[base + 1] = (y - (float)cy * GRIDF) / GRIDF;
        ldsOff[base + 2] = w / IMGF;
        ldsOff[base + 3] = h / IMGF;
    }
    __syncthreads();

    // ---- Phase C: per-cell loss terms ------------------------------------
    float c0 = 0.0f, c1 = 0.0f, c2 = 0.0f, c3 = 0.0f; // coord,obj,noobj,cls
    const int cell = lane64;
    if (cell < S_ * S_) {
        int last = -1;
        unsigned mask = 0u;
#pragma unroll
        for (int j = 0; j < NGT_; ++j) {
            if (ldsFlat[item * NGT_ + j] == cell) {
                last = j;                                 // max j wins (scatter-max)
                mask |= 1u << ldsLB[item * NGT_ + j];     // scatter-set labels
            }
        }
        const float2 cf = *(const float2*)(conf + ((size_t)n * (S_ * S_ * B_) + cell * B_));

        if (mask == 0u) {
            // no object in cell: only the no-object confidence penalty
            c2 = 0.5f * (cf.x * cf.x + cf.y * cf.y);
        } else {
            const int ob = (item * NGT_ + last) * 4;
            const float ox = ldsOff[ob + 0], oy = ldsOff[ob + 1];
            const float ow = ldsOff[ob + 2], oh = ldsOff[ob + 3];
            const float ltx = (float)(cell % S_) * GRIDF;
            const float lty = (float)(cell / S_) * GRIDF;

            // GT corners
            float gx = ox * GRIDF + ltx, gy = oy * GRIDF + lty;
            float ghw = ow * IMGF * 0.5f, ghh = oh * IMGF * 0.5f;
            float gx1 = clip01(gx - ghw), gy1 = clip01(gy - ghh);
            float gx2 = clip01(gx + ghw), gy2 = clip01(gy + ghh);
            float ga = (gx2 - gx1) * (gy2 - gy1);

            const float4* pb = (const float4*)(boxes + ((size_t)n * (S_ * S_ * B_ * 4) + cell * (B_ * 4)));
            float4 b0 = pb[0];
            float4 b1 = pb[1];
            float i0 = iou_box(b0.x, b0.y, b0.z, b0.w, ltx, lty, gx1, gy1, gx2, gy2, ga);
            float i1 = iou_box(b1.x, b1.y, b1.z, b1.w, ltx, lty, gx1, gy1, gx2, gy2, ga);
            const int best = (i1 > i0) ? 1 : 0;           // argmax, first-max on ties
            float4 pbB = best ? b1 : b0;
            float confB = best ? cf.y : cf.x;
            float confO = best ? cf.x : cf.y;

            float dx = pbB.x - ox;
            float dy = pbB.y - oy;
            float dw = sqrtf(pbB.z) - sqrtf(ow);
            float dh = sqrtf(pbB.w) - sqrtf(oh);
            c0 = 5.0f * (dx * dx + dy * dy + dw * dw + dh * dh);
            float e = confB - 1.0f;
            c1 = e * e;
            c2 = 0.5f * (confO * confO);

            // class loss (only read the 20 class probs when the cell is live)
            const float4* pc = (const float4*)(cls + ((size_t)n * (S_ * S_ * NC_) + cell * NC_));
            float acc = 0.0f;
#pragma unroll
            for (int q = 0; q < 5; ++q) {
                float4 v = pc[q];
                float t0 = v.x - (float)((mask >> (q * 4 + 0)) & 1u);
                float t1 = v.y - (float)((mask >> (q * 4 + 1)) & 1u);
                float t2 = v.z - (float)((mask >> (q * 4 + 2)) & 1u);
                float t3 = v.w - (float)((mask >> (q * 4 + 3)) & 1u);
                acc += t0 * t0 + t1 * t1 + t2 * t2 + t3 * t3;
            }
            c3 = acc;
        }
    }

    // ---- Phase D: deterministic block tree reduction ---------------------
    red[tid] = make_float4(c0, c1, c2, c3);
    __syncthreads();
#pragma unroll
    for (int s = THREADS_ / 2; s > 0; s >>= 1) {
        if (tid < s) {
            float4 a = red[tid], b = red[tid + s];
            red[tid] = make_float4(a.x + b.x, a.y + b.y, a.z + b.z, a.w + b.w);
        }
        __syncthreads();
    }
    if (tid == 0) {
        *(float4*)(partial + (size_t)blk * 4) = red[0];
    }
}

// ---------------------------------------------------------------------------
// Final reduction: column sums of P[nrows][4] via V_WMMA_F32_16X16X4_F32.
// A = ones(16x4), B holds 4 partial rows per issue (lanes 0-15: K=0/1 in
// v0/v1, lanes 16-31: K=2/3), C chains the accumulator. D[0][n] (VGPR0,
// lanes 0..3) ends up with the per-component sums of the wave's row range.
// ---------------------------------------------------------------------------
__global__ __launch_bounds__(THREADS_)
void yolo_reduce_kernel(const float* __restrict__ P, float* __restrict__ out, int nrows)
{
    __shared__ float wsum[8][4];
    const int tid  = threadIdx.x;
    const int wave = tid >> 5;
    const int lane = tid & 31;
    const int half = lane >> 4;
    const int ncol = lane & 15;

    const int rowsPerWave = nrows >> 3;          // 8 waves
    const int rbase = wave * rowsPerWave;

    v2f a; a.x = 1.0f; a.y = 1.0f;               // all-ones 16x4 A matrix
    v8f acc = {0.f, 0.f, 0.f, 0.f, 0.f, 0.f, 0.f, 0.f};

    for (int it = 0; it < rowsPerWave; it += 4) {
        const int r = rbase + it + half * 2;
        float b0 = 0.0f, b1 = 0.0f;
        if (ncol < 4) {
            b0 = P[(size_t)r * 4 + ncol];        // B[half*2 + 0][ncol]
            b1 = P[(size_t)(r + 1) * 4 + ncol];  // B[half*2 + 1][ncol]
        }
        v2f b; b.x = b0; b.y = b1;
        acc = __builtin_amdgcn_wmma_f32_16x16x4_f32(
            /*neg_a=*/false, a, /*neg_b=*/false, b,
            /*c_mod=*/(short)0, acc, /*reuse_a=*/false, /*reuse_b=*/false);
    }

    if (lane < 4) wsum[wave][lane] = acc[0];     // D[0][lane]
    __syncthreads();
    if (tid < 4) {
        float s = 0.0f;
#pragma unroll
        for (int w = 0; w < 8; ++w) s += wsum[w][tid];
        out[tid] = s;                            // (coord, obj, noobj, cls)
    }
}

extern "C" void kernel_launch(void* const* d_in, const int* in_sizes, int n_in,
                              void* d_out, int out_size, void* d_ws, size_t ws_size,
                              hipStream_t stream) {
    const float* conf  = (const float*)d_in[0];  // (N,7,7,2)
    const float* boxes = (const float*)d_in[1];  // (N,7,7,8)
    const float* cls   = (const float*)d_in[2];  // (N,7,7,20)
    const float* gt    = (const float*)d_in[3];  // (N,20,4)
    const int*   lbl   = (const int*)d_in[4];    // (N,20)

    const int N      = in_sizes[0] / (S_ * S_ * B_);   // 16384
    const int blocks = N / ITEMS_PER_BLOCK;            // 4096

    float* partial = (float*)d_ws;                     // blocks*4 floats (64 KB)

    yolo_main_kernel<<<blocks, THREADS_, 0, stream>>>(conf, boxes, cls, gt, lbl, partial);
    yolo_reduce_kernel<<<1, THREADS_, 0, stream>>>(partial, (float*)d_out, blocks);
}